// HiResPrecipNet_TEST_85057532330032
// MI455X (gfx1250) — compile-verified
//
#include <hip/hip_runtime.h>
#include <hip/hip_bf16.h>
#include <math.h>

#define N_LOW   5000
#define N_HIGH  50000
#define E_L2H   450000
#define E_HH    800000
#define SEQ     25
#define HIN     25

typedef __attribute__((ext_vector_type(2))) float v2f;
typedef __attribute__((ext_vector_type(8))) float v8f;

static inline int ceil_div(int a, int b) { return (a + b - 1) / b; }

// ---------------------------------------------------------------- fill
__global__ void fill_u32_kernel(unsigned* __restrict__ p, unsigned v, int n) {
  int i = blockIdx.x * blockDim.x + threadIdx.x;
  if (i < n) p[i] = v;
}

// ---------------------------------------------------------------- GRU layer
// One thread per node. Weights cached in LDS; h / hnew kept in per-thread LDS
// columns (no cross-thread sharing -> no barriers needed inside the time loop).
__global__ __launch_bounds__(128) void gru_layer_kernel(
    const float* __restrict__ x,    // [N, SEQ, HIN]
    const float* __restrict__ Wih,  // [3*HIN, HIN]
    const float* __restrict__ bih,  // [3*HIN]
    const float* __restrict__ Whh,  // [3*HIN, HIN]
    const float* __restrict__ bhh,  // [3*HIN]
    float* __restrict__ y,          // [N, SEQ, HIN]
    int N) {
  const int BS = 128;
  __shared__ float sWih[3 * HIN * HIN];
  __shared__ float sWhh[3 * HIN * HIN];
  __shared__ float sbih[3 * HIN];
  __shared__ float sbhh[3 * HIN];
  __shared__ float hS[HIN * BS];
  __shared__ float hnS[HIN * BS];

  const int tid = threadIdx.x;
  for (int i = tid; i < 3 * HIN * HIN; i += BS) { sWih[i] = Wih[i]; sWhh[i] = Whh[i]; }
  for (int i = tid; i < 3 * HIN; i += BS)       { sbih[i] = bih[i]; sbhh[i] = bhh[i]; }
  __syncthreads();

  const int node = blockIdx.x * BS + tid;
  if (node >= N) return;

  #pragma unroll
  for (int j = 0; j < HIN; ++j) hS[j * BS + tid] = 0.f;

  const float* xn = x + (size_t)node * SEQ * HIN;
  float* yn = y + (size_t)node * SEQ * HIN;

  for (int t = 0; t < SEQ; ++t) {
    float xt[HIN];
    #pragma unroll
    for (int k = 0; k < HIN; ++k) xt[k] = xn[t * HIN + k];

    for (int j = 0; j < HIN; ++j) {
      float gir = sbih[j], giz = sbih[HIN + j], gin = sbih[2 * HIN + j];
      float ghr = sbhh[j], ghz = sbhh[HIN + j], ghn = sbhh[2 * HIN + j];
      #pragma unroll
      for (int k = 0; k < HIN; ++k) {
        const float xv = xt[k];
        const float hv = hS[k * BS + tid];
        gir += sWih[j * HIN + k] * xv;
        giz += sWih[(HIN + j) * HIN + k] * xv;
        gin += sWih[(2 * HIN + j) * HIN + k] * xv;
        ghr += sWhh[j * HIN + k] * hv;
        ghz += sWhh[(HIN + j) * HIN + k] * hv;
        ghn += sWhh[(2 * HIN + j) * HIN + k] * hv;
      }
      const float r = 1.f / (1.f + __expf(-(gir + ghr)));
      const float z = 1.f / (1.f + __expf(-(giz + ghz)));
      const float n = tanhf(gin + r * ghn);
      hnS[j * BS + tid] = (1.f - z) * n + z * hS[j * BS + tid];
    }
    #pragma unroll
    for (int j = 0; j < HIN; ++j) {
      const float hv = hnS[j * BS + tid];
      hS[j * BS + tid] = hv;
      yn[t * HIN + j] = hv;
    }
  }
}

// ---------------------------------------------------------------- concat xz
__global__ void concat_xz_kernel(const float* __restrict__ z,      // [N,6]
                                 const float* __restrict__ land,   // [N,1]
                                 float* __restrict__ xz, int N) {
  int i = blockIdx.x * blockDim.x + threadIdx.x;
  if (i >= N) return;
  #pragma unroll
  for (int c = 0; c < 6; ++c) xz[i * 7 + c] = z[i * 6 + c];
  xz[i * 7 + 6] = land[i];
}

// ---------------------------------------------------------------- WMMA fp32 GEMM
// C[M,N] = A[M,K] @ W[N,K]^T + bias[N], optional ReLU.
// One wave per 16x16 output tile, K-loop in steps of 4 using
// V_WMMA_F32_16X16X4_F32.  VGPR layouts per CDNA5 ISA 7.12.2:
//   A 16x4 : lanes 0-15 hold K={0,1}, lanes 16-31 hold K={2,3}, M = lane&15
//   B 4x16 : lanes 0-15 hold K={0,1}, lanes 16-31 hold K={2,3}, N = lane&15
//   C 16x16: vgpr r holds M = r + 8*(lane>=16), N = lane&15
// Edge tiles: row/col indices are CLAMPED (not zero-filled) -> unconditional
// branch-free loads; garbage rows/cols only pollute outputs that the store
// guards drop. Only the K-tail iteration (K%4 != 0) is predicated.
__global__ __launch_bounds__(256) void wmma_gemm_nt_kernel(
    const float* __restrict__ A, const float* __restrict__ W,
    const float* __restrict__ bias, float* __restrict__ C,
    int M, int N, int K, int relu) {
  const int lane = threadIdx.x & 31;
  const int wave = (blockIdx.x * blockDim.x + threadIdx.x) >> 5;
  const int ntile = (N + 15) >> 4;
  const int mtile = (M + 15) >> 4;
  if (wave >= mtile * ntile) return;   // wave-uniform: EXEC stays all-ones
  const int tm = wave / ntile;
  const int tn = wave % ntile;

  const int row = tm * 16 + (lane & 15);
  const int col = tn * 16 + (lane & 15);
  const int rowc = (row < M) ? row : (M - 1);
  const int colc = (col < N) ? col : (N - 1);
  const int kh = (lane >> 4) * 2;  // 0 or 2

  const float* __restrict__ Ar = A + (size_t)rowc * K + kh;
  const float* __restrict__ Wc = W + (size_t)colc * K + kh;

  v8f acc = {0.f, 0.f, 0.f, 0.f, 0.f, 0.f, 0.f, 0.f};
  const int K4 = K & ~3;
  int k0 = 0;
  for (; k0 < K4; k0 += 4) {
    v2f a, b;
    a.x = Ar[k0];
    a.y = Ar[k0 + 1];
    b.x = Wc[k0];
    b.y = Wc[k0 + 1];
    acc = __builtin_amdgcn_wmma_f32_16x16x4_f32(
        false, a, false, b, (short)0, acc, false, false);
  }
  if (k0 < K) {  // K-tail (at most one iteration)
    const int ka = k0 + kh;
    v2f a = {0.f, 0.f}, b = {0.f, 0.f};
    if (ka < K)     { a.x = Ar[k0];     b.x = Wc[k0]; }
    if (ka + 1 < K) { a.y = Ar[k0 + 1]; b.y = Wc[k0 + 1]; }
    acc = __builtin_amdgcn_wmma_f32_16x16x4_f32(
        false, a, false, b, (short)0, acc, false, false);
  }

  const float bcol = (col < N) ? bias[col] : 0.f;
  const int rbase = tm * 16 + ((lane >> 4) << 3);
  #pragma unroll
  for (int r = 0; r < 8; ++r) {
    const int rr = rbase + r;
    if (rr < M && col < N) {
      float v = acc[r] + bcol;
      if (relu) v = fmaxf(v, 0.f);
      C[(size_t)rr * N + col] = v;
    }
  }
}

// ---------------------------------------------------------------- GAT edge kernels
__device__ __forceinline__ void edge_sd(int e, int nE,
                                        const int* __restrict__ src,
                                        const int* __restrict__ dst,
                                        int& s, int& d) {
  if (e < nE) { s = src[e]; d = dst[e]; }
  else        { s = e - nE; d = e - nE; }  // self loops
}

// monotonic float <-> uint encoding for atomicMax-based segment max
__device__ __forceinline__ unsigned enc_f32(float f) {
  unsigned b = __float_as_uint(f);
  return (b & 0x80000000u) ? ~b : (b | 0x80000000u);
}
__device__ __forceinline__ float dec_f32(unsigned e) {
  unsigned b = (e & 0x80000000u) ? (e & 0x7FFFFFFFu) : ~e;
  return __uint_as_float(b);
}

__global__ void edge_logit_max_kernel(
    const float* __restrict__ xl, const float* __restrict__ xr,
    const float* __restrict__ att,
    const int* __restrict__ src, const int* __restrict__ dst,
    int nE, int nSelf, int H, int C,
    float* __restrict__ elogit, unsigned* __restrict__ mEnc) {
  const int idx = blockIdx.x * blockDim.x + threadIdx.x;
  const int total = (nE + nSelf) * H;
  if (idx >= total) return;
  const int e = idx / H, h = idx % H;
  int s, d; edge_sd(e, nE, src, dst, s, d);
  // rows are 256B-aligned (h*C multiples of 64 floats, 16B-aligned buffers)
  const float4* __restrict__ pl = (const float4*)(xl + (size_t)s * H * C + h * C);
  const float4* __restrict__ pr = (const float4*)(xr + (size_t)d * H * C + h * C);
  const float4* __restrict__ pa = (const float4*)(att + h * C);
  float acc = 0.f;
  #pragma unroll 4
  for (int c4 = 0; c4 < C / 4; ++c4) {
    const float4 l = pl[c4];
    const float4 r = pr[c4];
    const float4 a = pa[c4];
    float q0 = l.x + r.x; q0 = (q0 > 0.f) ? q0 : 0.2f * q0;
    float q1 = l.y + r.y; q1 = (q1 > 0.f) ? q1 : 0.2f * q1;
    float q2 = l.z + r.z; q2 = (q2 > 0.f) ? q2 : 0.2f * q2;
    float q3 = l.w + r.w; q3 = (q3 > 0.f) ? q3 : 0.2f * q3;
    acc += q0 * a.x + q1 * a.y + q2 * a.z + q3 * a.w;
  }
  elogit[idx] = acc;
  atomicMax(&mEnc[(size_t)d * H + h], enc_f32(acc));
}

__global__ void edge_softmax_kernel(
    float* __restrict__ elogit_pexp,   // in: logits, out: exp(e - m)
    const int* __restrict__ dst,
    int nE, int nSelf, int H,
    const unsigned* __restrict__ mEnc,
    float* __restrict__ denom, float* __restrict__ deg) {
  const int idx = blockIdx.x * blockDim.x + threadIdx.x;
  const int total = (nE + nSelf) * H;
  if (idx >= total) return;
  const int e = idx / H, h = idx % H;
  const int d = (e < nE) ? dst[e] : (e - nE);
  const float m = dec_f32(mEnc[(size_t)d * H + h]);
  const float p = __expf(elogit_pexp[idx] - m);
  elogit_pexp[idx] = p;
  atomicAdd(&denom[(size_t)d * H + h], p);
  if (h == 0) atomicAdd(&deg[d], 1.f);
}

__global__ void edge_aggregate_kernel(
    const float* __restrict__ xl, const float* __restrict__ pexp,
    const float* __restrict__ denom,
    const int* __restrict__ src, const int* __restrict__ dst,
    int nE, int nSelf, int H, int C,
    float* __restrict__ sAcc) {
  const int idx = blockIdx.x * blockDim.x + threadIdx.x;
  const int HC = H * C;
  const int total = (nE + nSelf) * HC;
  if (idx >= total) return;
  const int e = idx / HC, hc = idx % HC, h = hc / C;
  int s, d; edge_sd(e, nE, src, dst, s, d);
  const float alpha = pexp[(size_t)e * H + h] / denom[(size_t)d * H + h];
  atomicAdd(&sAcc[(size_t)d * HC + hc], alpha * xl[(size_t)s * HC + hc]);
}

__global__ void gat_finalize_kernel(
    const float* __restrict__ sAcc, const float* __restrict__ deg,
    const float* __restrict__ bias, float* __restrict__ out,
    int N, int HC, int relu) {
  const int idx = blockIdx.x * blockDim.x + threadIdx.x;
  if (idx >= N * HC) return;
  const int n = idx / HC, c = idx % HC;
  float dg = deg[n];
  dg = (dg > 1.f) ? dg : 1.f;
  float v = sAcc[idx] / dg + bias[c];
  if (relu) v = fmaxf(v, 0.f);
  out[idx] = v;
}

// ---------------------------------------------------------------- BatchNorm (training mode)
__global__ void bn_stats_kernel(const float* __restrict__ x, int N, int D,
                                float* __restrict__ sum, float* __restrict__ sumsq) {
  const int c = threadIdx.x;             // blockDim.x == D
  float s = 0.f, q = 0.f;
  for (int r = blockIdx.x; r < N; r += gridDim.x) {
    const float v = x[(size_t)r * D + c];
    s += v; q += v * v;
  }
  atomicAdd(&sum[c], s);
  atomicAdd(&sumsq[c], q);
}

__global__ void bn_apply_kernel(float* __restrict__ x,
                                const float* __restrict__ sum,
                                const float* __restrict__ sumsq,
                                const float* __restrict__ g,
                                const float* __restrict__ b,
                                int N, int D, int relu) {
  const int idx = blockIdx.x * blockDim.x + threadIdx.x;
  if (idx >= N * D) return;
  const int c = idx % D;
  const float invN = 1.f / (float)N;
  const float mu = sum[c] * invN;
  const float var = sumsq[c] * invN - mu * mu;
  float v = (x[idx] - mu) * rsqrtf(var + 1e-5f) * g[c] + b[c];
  if (relu) v = fmaxf(v, 0.f);
  x[idx] = v;
}

// ---------------------------------------------------------------- host helpers
static void gemm(const float* A, const float* W, const float* bias, float* C,
                 int M, int N, int K, int relu, hipStream_t stream) {
  const int mt = (M + 15) / 16, nt = (N + 15) / 16;
  const long waves = (long)mt * nt;
  const int blocks = (int)((waves + 7) / 8);  // 8 wave32 per 256-thread block
  wmma_gemm_nt_kernel<<<blocks, 256, 0, stream>>>(A, W, bias, C, M, N, K, relu);
}

static void run_gat(const float* xl, const float* xr, const float* att, const float* bias,
                    const int* src, const int* dst, int nE, int nSelf,
                    int Ndst, int H, int C, float* out, int relu,
                    float* sAcc, float* pexp, unsigned* mEnc, float* denom, float* deg,
                    hipStream_t stream) {
  const int nEdge = nE + nSelf;
  const int totalEH = nEdge * H;
  const int ndh = Ndst * H;
  const int ndhc = Ndst * H * C;
  fill_u32_kernel<<<ceil_div(ndh, 256), 256, 0, stream>>>(mEnc, 0u, ndh);
  fill_u32_kernel<<<ceil_div(ndh, 256), 256, 0, stream>>>((unsigned*)denom, 0u, ndh);
  fill_u32_kernel<<<ceil_div(Ndst, 256), 256, 0, stream>>>((unsigned*)deg, 0u, Ndst);
  fill_u32_kernel<<<ceil_div(ndhc, 256), 256, 0, stream>>>((unsigned*)sAcc, 0u, ndhc);
  edge_logit_max_kernel<<<ceil_div(totalEH, 256), 256, 0, stream>>>(
      xl, xr, att, src, dst, nE, nSelf, H, C, pexp, mEnc);
  edge_softmax_kernel<<<ceil_div(totalEH, 256), 256, 0, stream>>>(
      pexp, dst, nE, nSelf, H, mEnc, denom, deg);
  const int totalEHC = nEdge * H * C;  // max 108.8M < 2^31
  edge_aggregate_kernel<<<ceil_div(totalEHC, 256), 256, 0, stream>>>(
      xl, pexp, denom, src, dst, nE, nSelf, H, C, sAcc);
  gat_finalize_kernel<<<ceil_div(ndhc, 256), 256, 0, stream>>>(
      sAcc, deg, bias, out, Ndst, H * C, relu);
}

static void run_bn(float* x, const float* g, const float* b, int N, int D, int relu,
                   float* stats, hipStream_t stream) {
  fill_u32_kernel<<<1, 256, 0, stream>>>((unsigned*)stats, 0u, 2 * D);
  bn_stats_kernel<<<256, D, 0, stream>>>(x, N, D, stats, stats + D);
  bn_apply_kernel<<<ceil_div(N * D, 256), 256, 0, stream>>>(x, stats, stats + D, g, b, N, D, relu);
}

// ---------------------------------------------------------------- entry
extern "C" void kernel_launch(void* const* d_in, const int* in_sizes, int n_in,
                              void* d_out, int out_size, void* d_ws, size_t ws_size,
                              hipStream_t stream) {
  // ---- inputs (setup_inputs dict order, params flattened in insertion order)
  const float* x_low    = (const float*)d_in[0];
  const float* z_std    = (const float*)d_in[1];
  const float* land_std = (const float*)d_in[2];
  const int*   edge_l2h = (const int*)d_in[3];   // [2, E_L2H]
  const int*   edge_hh  = (const int*)d_in[4];   // [2, E_HH]
  // gru: 2 x (Wih, bih, Whh, bhh) -> 5..12
  const float* gruW[2][4];
  for (int l = 0; l < 2; ++l)
    for (int k = 0; k < 4; ++k) gruW[l][k] = (const float*)d_in[5 + l * 4 + k];
  const float* denseW = (const float*)d_in[13];  // [128, 625]
  const float* denseB = (const float*)d_in[14];
  // down gat: Wl, bl, Wr, br, att, bias -> 15..20
  const float* dWl = (const float*)d_in[15];
  const float* dbl = (const float*)d_in[16];
  const float* dWr = (const float*)d_in[17];
  const float* dbr = (const float*)d_in[18];
  const float* datt = (const float*)d_in[19];
  const float* dbias = (const float*)d_in[20];
  // bn: 5 x (g, b) -> 21..30
  const float* bnG[5]; const float* bnB[5];
  for (int i = 0; i < 5; ++i) { bnG[i] = (const float*)d_in[21 + 2 * i]; bnB[i] = (const float*)d_in[22 + 2 * i]; }
  // gats: 5 x (Wl, bl, Wr, br, att, bias) -> 31..60
  const float* gWl[5]; const float* gbl[5]; const float* gWr[5];
  const float* gbr[5]; const float* gatt[5]; const float* gbias[5];
  for (int i = 0; i < 5; ++i) {
    gWl[i]   = (const float*)d_in[31 + 6 * i + 0];
    gbl[i]   = (const float*)d_in[31 + 6 * i + 1];
    gWr[i]   = (const float*)d_in[31 + 6 * i + 2];
    gbr[i]   = (const float*)d_in[31 + 6 * i + 3];
    gatt[i]  = (const float*)d_in[31 + 6 * i + 4];
    gbias[i] = (const float*)d_in[31 + 6 * i + 5];
  }
  // pred: 3 x (W, b) -> 61..66
  const float* pW[3]; const float* pB[3];
  for (int i = 0; i < 3; ++i) { pW[i] = (const float*)d_in[61 + 2 * i]; pB[i] = (const float*)d_in[62 + 2 * i]; }

  // ---- workspace carve (floats)
  float* wsf = (float*)d_ws;
  size_t off = 0;
  auto alloc = [&](size_t n) { float* p = wsf + off; off += n; return p; };
  float* seq1  = alloc((size_t)N_LOW * SEQ * HIN);     // 3.125M
  float* seq2  = alloc((size_t)N_LOW * SEQ * HIN);     // 3.125M
  float* xz    = alloc((size_t)N_HIGH * 8);            // keep 16B row alignment slack
  float* enc   = alloc((size_t)N_LOW * 128);
  float* xl    = alloc((size_t)N_HIGH * 128);
  float* xr    = alloc((size_t)N_HIGH * 128);
  float* xA    = alloc((size_t)N_HIGH * 128);
  float* xB    = alloc((size_t)N_HIGH * 128);
  float* sAcc  = alloc((size_t)N_HIGH * 128);
  float* pexp  = alloc((size_t)(E_HH + N_HIGH) * 2);
  unsigned* mEnc = (unsigned*)alloc((size_t)N_HIGH * 2);
  float* denom = alloc((size_t)N_HIGH * 2);
  float* deg   = alloc((size_t)N_HIGH);
  float* stats = alloc(256);
  (void)ws_size; (void)in_sizes; (void)n_in; (void)out_size;

  // ---- 1. two stacked GRU layers over x_low
  gru_layer_kernel<<<ceil_div(N_LOW, 128), 128, 0, stream>>>(
      x_low, gruW[0][0], gruW[0][1], gruW[0][2], gruW[0][3], seq1, N_LOW);
  gru_layer_kernel<<<ceil_div(N_LOW, 128), 128, 0, stream>>>(
      seq1, gruW[1][0], gruW[1][1], gruW[1][2], gruW[1][3], seq2, N_LOW);

  // ---- 2. dense encoder: enc = relu(seq2.reshape(N,625) @ Wd^T + bd)
  gemm(seq2, denseW, denseB, enc, N_LOW, 128, SEQ * HIN, /*relu=*/1, stream);

  // ---- 3. xz = concat(z_std, land_std)
  concat_xz_kernel<<<ceil_div(N_HIGH, 256), 256, 0, stream>>>(z_std, land_std, xz, N_HIGH);

  // ---- 4. down GATv2 (low -> high), H=1, C=64
  gemm(enc, dWl, dbl, xl, N_LOW, 64, 128, 0, stream);
  gemm(xz,  dWr, dbr, xr, N_HIGH, 64, 7,  0, stream);
  run_gat(xl, xr, datt, dbias,
          edge_l2h /*src row*/, edge_l2h + E_L2H /*dst row*/, E_L2H, /*nSelf=*/0,
          N_HIGH, /*H=*/1, /*C=*/64, xA, /*relu=*/0,
          sAcc, pexp, mEnc, denom, deg, stream);

  // ---- 5. five GATv2 layers over high-high edges (+ self loops)
  const int din[5]  = {64, 128, 128, 128, 128};
  const int Hh[5]   = {2, 2, 2, 2, 1};
  float* cur = xA;
  float* nxt = xB;
  for (int i = 0; i < 5; ++i) {
    run_bn(cur, bnG[i], bnB[i], N_HIGH, din[i], /*relu=*/(i > 0) ? 1 : 0, stats, stream);
    const int hc = Hh[i] * 64;
    gemm(cur, gWl[i], gbl[i], xl, N_HIGH, hc, din[i], 0, stream);
    gemm(cur, gWr[i], gbr[i], xr, N_HIGH, hc, din[i], 0, stream);
    run_gat(xl, xr, gatt[i], gbias[i],
            edge_hh /*src*/, edge_hh + E_HH /*dst*/, E_HH, /*nSelf=*/N_HIGH,
            N_HIGH, Hh[i], 64, nxt, /*relu=*/(i == 4) ? 1 : 0,  // final relu folded in
            sAcc, pexp, mEnc, denom, deg, stream);
    float* t = cur; cur = nxt; nxt = t;
  }

  // ---- 6. prediction MLP: 64 -> 64(relu) -> 32(relu) -> 1
  gemm(cur, pW[0], pB[0], xl, N_HIGH, 64, 64, 1, stream);
  gemm(xl,  pW[1], pB[1], xr, N_HIGH, 32, 64, 1, stream);
  gemm(xr,  pW[2], pB[2], (float*)d_out, N_HIGH, 1, 32, 0, stream);
}